// ScaledDotProductAttention_34849364639933
// MI455X (gfx1250) — compile-verified
//
#include <hip/hip_runtime.h>

// Problem constants (from reference): q,k,v,pos are [BH=16, L=2048, D=64] fp32.
// out = [BH,L,D] fp32 ; attn = [BH,L,L] fp32, concatenated flat in d_out.
#define BH 16
#define L  2048
#define D  64
#define ROWS 16            // rows of the S stripe handled per workgroup
#define LT (L / 16)        // 128 column tiles of 16
#define NCHUNK (LT / 8)    // 16 P-chunks of 128 columns
#define SSTRIDE (L + 4)    // padded LDS stride for the S stripe (floats)
#define PSTRIDE 68         // padded LDS stride for P-stage / QM (floats)
#define NW 8               // waves per block (256 threads, wave32)

typedef __attribute__((ext_vector_type(2))) float v2f;
typedef __attribute__((ext_vector_type(8))) float v8f;
typedef int v4i_b __attribute__((vector_size(16))); // matches builtin proto

#define AS1 __attribute__((address_space(1)))
#define AS3 __attribute__((address_space(3)))

#if __has_builtin(__builtin_amdgcn_global_load_async_to_lds_b128)
#define HAVE_ASYNC_LDS 1
#else
#define HAVE_ASYNC_LDS 0
#endif

// 16B global -> LDS copy. Async (ASYNCcnt-tracked, bypasses VGPRs) when the
// gfx1250 builtin is available, synchronous float4 otherwise.
__device__ __forceinline__ void cp16_g2l(const float* __restrict__ g,
                                         float* __restrict__ l) {
#if HAVE_ASYNC_LDS
  __builtin_amdgcn_global_load_async_to_lds_b128(
      (AS1 v4i_b*)g, (AS3 v4i_b*)l, 0, 0);
#else
  *(float4*)l = *(const float4*)g;
#endif
}

__device__ __forceinline__ void wait_async_lds() {
#if HAVE_ASYNC_LDS
#if __has_builtin(__builtin_amdgcn_s_wait_asynccnt)
  __builtin_amdgcn_s_wait_asynccnt(0);
#else
  asm volatile("s_wait_asynccnt 0x0" ::: "memory");
#endif
#endif
}

// D = A(16x4) * B(4x16) + C, all fp32 (V_WMMA_F32_16X16X4_F32).
__device__ __forceinline__ v8f wmma4(v2f a, v2f b, v8f c) {
  return __builtin_amdgcn_wmma_f32_16x16x4_f32(
      /*neg_a=*/false, a, /*neg_b=*/false, b,
      /*c_mod=*/(short)0, c, /*reuse_a=*/false, /*reuse_b=*/false);
}

// ---------------------------------------------------------------------------
// Kernel 1: M[bh] = K[bh]^T Q[bh] / t^2   (64x64 per batch-head)
// 16 waves per block; wave w owns 16x16 tile (w>>2, w&3); K-reduction = 2048.
// ---------------------------------------------------------------------------
__global__ __launch_bounds__(512) void dpa_mmat_kernel(
    const float* __restrict__ Kin, const float* __restrict__ Qin,
    float* __restrict__ Mout) {
  const int bh  = blockIdx.x;
  const int tid = threadIdx.x;
  const int w   = tid >> 5;
  const int l   = tid & 31;
  const int hl  = l >> 4;        // 0: lanes 0-15, 1: lanes 16-31
  const int l16 = l & 15;
  const int ti  = (w >> 2) * 16; // M row-tile base (index i over D)
  const int tj  = (w & 3) * 16;  // M col-tile base (index j over D)

  const float* Kb = Kin + (size_t)bh * L * D;
  const float* Qb = Qin + (size_t)bh * L * D;

  v8f acc0 = {};
  v8f acc1 = {};
#pragma unroll 2
  for (int k = 0; k < L; k += 8) {
    {
      const int k0 = k + hl * 2; // lanes 16-31 carry K=2,3 of the step
      v2f a, b;
      a.x = Kb[(size_t)(k0 + 0) * D + ti + l16]; // A[m][c] = K[k+c][ti+m]
      a.y = Kb[(size_t)(k0 + 1) * D + ti + l16];
      b.x = Qb[(size_t)(k0 + 0) * D + tj + l16]; // B[c][n] = Q[k+c][tj+n]
      b.y = Qb[(size_t)(k0 + 1) * D + tj + l16];
      acc0 = wmma4(a, b, acc0);
    }
    {
      const int k0 = k + 4 + hl * 2;
      v2f a, b;
      a.x = Kb[(size_t)(k0 + 0) * D + ti + l16];
      a.y = Kb[(size_t)(k0 + 1) * D + ti + l16];
      b.x = Qb[(size_t)(k0 + 0) * D + tj + l16];
      b.y = Qb[(size_t)(k0 + 1) * D + tj + l16];
      acc1 = wmma4(a, b, acc1);
    }
  }
  acc0 = acc0 + acc1;

  const float scale = 1.0f / 64.0f; // 1/t^2, t = sqrt(64) = 8
  float* Mb = Mout + (size_t)bh * D * D;
#pragma unroll
  for (int v = 0; v < 8; ++v) {
    const int m = v + hl * 8; // D-frag: VGPR v -> row v (lanes<16) / v+8
    Mb[(ti + m) * D + tj + l16] = acc0[v] * scale;
  }
}

// ---------------------------------------------------------------------------
// Kernel 2: fused  S = (Q M) P^T  ->  softmax  ->  attn out + O = P V
// grid = (L/ROWS, BH), block = 256 threads (8 waves). Full 16x2048 S stripe
// lives in LDS (CDNA5 WGP has 320KB LDS); attention matrix is written to HBM
// exactly once, already normalized. P tiles are staged through a
// double-buffered LDS region filled with async global->LDS loads so the next
// chunk's fill overlaps the current chunk's WMMA work.
// ---------------------------------------------------------------------------
__global__ __launch_bounds__(256) void dpa_attn_kernel(
    const float* __restrict__ Qin, const float* __restrict__ Pin,
    const float* __restrict__ Vin, const float* __restrict__ Mmat,
    float* __restrict__ out, float* __restrict__ attn) {
  extern __shared__ float smem[];
  float* sQM   = smem;                         // ROWS * PSTRIDE
  float* sS    = sQM + ROWS * PSTRIDE;         // ROWS * SSTRIDE
  float* sRed  = sS + ROWS * SSTRIDE;          // 16 * 16
  float* sStat = sRed + 256;                   // 16
  float* sO    = sStat + 16;                   // NW * ROWS * D
  float* sP0   = sO + NW * ROWS * D;           // 128 * PSTRIDE (P stage A)
  float* sP1   = sP0 + 128 * PSTRIDE;          // 128 * PSTRIDE (P stage B)

  const int bh  = blockIdx.y;
  const int r0  = blockIdx.x * ROWS;
  const int tid = threadIdx.x;
  const int w   = tid >> 5;
  const int l   = tid & 31;
  const int hl  = l >> 4;
  const int l16 = l & 15;

  const float* Qb = Qin + ((size_t)bh * L + r0) * D; // our 16 Q rows
  const float* Pb = Pin + (size_t)bh * L * D;
  const float* Vb = Vin + (size_t)bh * L * D;
  const float* Mb = Mmat + (size_t)bh * D * D;

  // ---- Phase 0: QM rows = Q[r0..r0+16) @ M  (waves 0-3, one 16x16 tile
  // each); meanwhile kick off the async fill of P chunk 0.
  {
    for (int idx = tid; idx < 128 * (D / 4); idx += 256) {
      const int col = idx >> 4;
      const int pos = (idx & 15) << 2;
      cp16_g2l(&Pb[(size_t)col * D + pos], &sP0[col * PSTRIDE + pos]);
    }
  }
  if (w < 4) {
    const int j0 = w * 16;
    v8f acc = {};
#pragma unroll
    for (int k = 0; k < D; k += 4) {
      const int c0 = k + hl * 2;
      v2f a, b;
      a.x = Qb[l16 * D + c0 + 0];            // A[m][c] = Q[r0+m][k+c]
      a.y = Qb[l16 * D + c0 + 1];
      b.x = Mb[(c0 + 0) * D + j0 + l16];     // B[c][n] = M[k+c][j0+n]
      b.y = Mb[(c0 + 1) * D + j0 + l16];
      acc = wmma4(a, b, acc);
    }
#pragma unroll
    for (int v = 0; v < 8; ++v)
      sQM[(v + hl * 8) * PSTRIDE + j0 + l16] = acc[v];
  }
  wait_async_lds();
  __syncthreads();

  // A-fragments of QM for the S GEMM (kept in registers for the whole
  // stripe). Consecutive pair -> single ds_load_b64.
  v2f aS[16];
#pragma unroll
  for (int kk = 0; kk < 16; ++kk)
    aS[kk] = *(const v2f*)&sQM[l16 * PSTRIDE + kk * 4 + hl * 2];

  // ---- Phase 1: S stripe = QM @ P^T, 128 col-tiles; double-buffered P
  // stage, wave w computes tile cc*8+w of each chunk.
  for (int cc = 0; cc < NCHUNK; ++cc) {
    float* cur = (cc & 1) ? sP1 : sP0;
    float* nxt = (cc & 1) ? sP0 : sP1;
    if (cc + 1 < NCHUNK) { // issue async fill of the next chunk
      for (int idx = tid; idx < 128 * (D / 4); idx += 256) {
        const int col = idx >> 4;
        const int pos = (idx & 15) << 2;
        cp16_g2l(&Pb[(size_t)((cc + 1) * 128 + col) * D + pos],
                 &nxt[col * PSTRIDE + pos]);
      }
    }

    const int cbase = (cc * NW + w) * 16;
    v8f acc = {};
#pragma unroll
    for (int kk = 0; kk < 16; ++kk) {
      // B[c][n] = P[cbase+n][kk*4+c], read transposed from the LDS stage.
      const v2f b = *(const v2f*)&cur[(w * 16 + l16) * PSTRIDE + kk * 4 + hl * 2];
      acc = wmma4(aS[kk], b, acc);
    }
#pragma unroll
    for (int v = 0; v < 8; ++v)
      sS[(v + hl * 8) * SSTRIDE + cbase + l16] = acc[v];

    wait_async_lds();
    __syncthreads();
  }

  // ---- Phase 2: exact softmax on the LDS stripe + single coalesced write of
  // the normalized attention matrix to HBM.
  {
    const int row = tid >> 4;  // 0..15
    const int c0  = tid & 15;
    float* Srow = &sS[row * SSTRIDE];

    // warm V for phase 3 (global_prefetch_b8)
    __builtin_prefetch(Vb + (size_t)tid * 32, 0, 0);

    float m = -__builtin_inff();
    for (int c = c0; c < L; c += 16) m = fmaxf(m, Srow[c]);
    sRed[row * 16 + c0] = m;
    __syncthreads();
    if (c0 == 0) {
      float mm = sRed[row * 16];
#pragma unroll
      for (int i = 1; i < 16; ++i) mm = fmaxf(mm, sRed[row * 16 + i]);
      sStat[row] = mm;
    }
    __syncthreads();
    const float rowmax = sStat[row];
    float ssum = 0.0f;
    for (int c = c0; c < L; c += 16) {
      const float e = __expf(Srow[c] - rowmax);
      Srow[c] = e;
      ssum += e;
    }
    sRed[row * 16 + c0] = ssum;
    __syncthreads();
    if (c0 == 0) {
      float t = 0.0f;
#pragma unroll
      for (int i = 0; i < 16; ++i) t += sRed[row * 16 + i];
      sStat[row] = 1.0f / t;
    }
    __syncthreads();
    const float inv = sStat[row];
    for (int c = c0; c < L; c += 16) Srow[c] *= inv;
    __syncthreads();

    // coalesced float4 writeback of normalized probabilities
    for (int idx = tid; idx < ROWS * (L / 4); idx += 256) {
      const int r  = idx / (L / 4);
      const int c4 = (idx % (L / 4)) * 4;
      const float4 pv = *(const float4*)&sS[r * SSTRIDE + c4];
      *(float4*)&attn[((size_t)bh * L + r0 + r) * L + c4] = pv;
    }
    __syncthreads();
  }

  // ---- Phase 3: O = P_prob @ V, probabilities read from LDS, per-wave
  // partial accumulators reduced through LDS.
  v8f accO[4];
#pragma unroll
  for (int jg = 0; jg < 4; ++jg) accO[jg] = (v8f){};

  for (int ct = w; ct < LT; ct += NW) {
    const int cbase = ct * 16;
#pragma unroll
    for (int kk = 0; kk < 4; ++kk) {
      // A[m][c] = P_prob[m][cbase + kk*4 + c] (single ds_load_b64)
      const v2f a = *(const v2f*)&sS[l16 * SSTRIDE + cbase + kk * 4 + hl * 2];
#pragma unroll
      for (int jg = 0; jg < 4; ++jg) {
        v2f b; // B[c][n] = V[cbase + kk*4 + c][jg*16 + n]
        b.x = Vb[(size_t)(cbase + kk * 4 + hl * 2 + 0) * D + jg * 16 + l16];
        b.y = Vb[(size_t)(cbase + kk * 4 + hl * 2 + 1) * D + jg * 16 + l16];
        accO[jg] = wmma4(a, b, accO[jg]);
      }
    }
  }

#pragma unroll
  for (int jg = 0; jg < 4; ++jg)
#pragma unroll
    for (int v = 0; v < 8; ++v)
      sO[w * (ROWS * D) + (v + hl * 8) * D + jg * 16 + l16] = accO[jg][v];
  __syncthreads();

  // reduce 8 wave-partials and store O rows (one float4 per thread)
  {
    const int idx = tid; // ROWS*D/4 == 256 == blockDim
    float4 s = {0.f, 0.f, 0.f, 0.f};
#pragma unroll
    for (int ww = 0; ww < NW; ++ww) {
      const float4 p = *(const float4*)&sO[ww * (ROWS * D) + idx * 4];
      s.x += p.x; s.y += p.y; s.z += p.z; s.w += p.w;
    }
    const int row = (idx * 4) / D;
    const int col = (idx * 4) % D;
    *(float4*)&out[((size_t)bh * L + r0 + row) * D + col] = s;
  }
}

// ---------------------------------------------------------------------------
extern "C" void kernel_launch(void* const* d_in, const int* in_sizes, int n_in,
                              void* d_out, int out_size, void* d_ws,
                              size_t ws_size, hipStream_t stream) {
  const float* q = (const float*)d_in[0];
  const float* k = (const float*)d_in[1];
  const float* v = (const float*)d_in[2];
  const float* p = (const float*)d_in[3];

  float* out  = (float*)d_out;                  // [BH, L, D]
  float* attn = out + (size_t)BH * L * D;       // [BH, L, L]
  float* Mws  = (float*)d_ws;                   // [BH, D, D] = 256 KB scratch

  // Kernel 1: M = K^T Q / t^2 per batch-head.
  dpa_mmat_kernel<<<dim3(BH), dim3(512), 0, stream>>>(k, q, Mws);

  // Kernel 2: fused QM / S-stripe / softmax / attention write / O = P V.
  constexpr size_t SMEM_FLOATS = (size_t)ROWS * PSTRIDE       // sQM
                               + (size_t)ROWS * SSTRIDE       // sS stripe
                               + 256 + 16                     // sRed, sStat
                               + (size_t)NW * ROWS * D        // sO partials
                               + (size_t)2 * 128 * PSTRIDE;   // sP0/sP1 stage
  const size_t shmem = SMEM_FLOATS * sizeof(float);           // ~234 KB (<320)
  (void)hipFuncSetAttribute((const void*)dpa_attn_kernel,
                            hipFuncAttributeMaxDynamicSharedMemorySize,
                            (int)shmem);
  dpa_attn_kernel<<<dim3(L / ROWS, BH), dim3(256), shmem, stream>>>(
      q, p, v, Mws, out, attn);
}